// SGCEncoder_66915590472501
// MI455X (gfx1250) — compile-verified
//
#include <hip/hip_runtime.h>

typedef float v2f __attribute__((ext_vector_type(2)));
typedef float v8f __attribute__((ext_vector_type(8)));

#define D_IN  256
#define D_HID 256
#define D_OUT 128

// ---------------------------------------------------------------------------
// Degree / normalization
// ---------------------------------------------------------------------------
__global__ void deg_init_kernel(float* __restrict__ deg, int n) {
  int i = blockIdx.x * blockDim.x + threadIdx.x;
  if (i < n) deg[i] = 1.0f;  // self-loop contribution
}

__global__ void deg_scatter_kernel(const int* __restrict__ dst,
                                   float* __restrict__ deg, int E) {
  int e = blockIdx.x * blockDim.x + threadIdx.x;
  if (e < E) atomicAdd(&deg[dst[e]], 1.0f);
}

__global__ void dinv_kernel(float* __restrict__ deg, int n) {
  int i = blockIdx.x * blockDim.x + threadIdx.x;
  if (i < n) deg[i] = rsqrtf(fmaxf(deg[i], 1e-12f));
}

// ---------------------------------------------------------------------------
// GEMM1: y[N,256] = x[N,256] @ W1[256,256], fp32 WMMA 16x16x4.
// One wave: 16-row tile x 4 col-tiles (64 cols), A fragment reused 4x.
// ---------------------------------------------------------------------------
__global__ void gemm_xw1_kernel(const float* __restrict__ x,
                                const float* __restrict__ W1,
                                float* __restrict__ y, int rowTiles) {
  const int wave = (blockIdx.x * blockDim.x + threadIdx.x) >> 5;
  const int lane = threadIdx.x & 31;
  const int colGroups = D_HID / 64;   // 4
  const int rt = wave / colGroups;
  const int cg = wave % colGroups;
  if (rt >= rowTiles) return;

  const int m0   = rt * 16;
  const int n0   = cg * 64;
  const int half = lane >> 4;   // 0: K pair {0,1}; 1: K pair {2,3}
  const int l16  = lane & 15;

  v8f acc[4] = {};
  const float* arow = x + (size_t)(m0 + l16) * D_IN + 2 * half;

  for (int k = 0; k < D_IN; k += 4) {
    if (k + 64 < D_IN) __builtin_prefetch(arow + k + 64, 0, 1);
    v2f a;
    a.x = arow[k];
    a.y = arow[k + 1];
    const float* bp = W1 + (size_t)(k + 2 * half) * D_HID + n0 + l16;
#pragma unroll
    for (int j = 0; j < 4; ++j) {
      v2f b;
      b.x = bp[j * 16];
      b.y = bp[j * 16 + D_HID];
      acc[j] = __builtin_amdgcn_wmma_f32_16x16x4_f32(
          false, a, false, b, (short)0, acc[j], false, false);
    }
  }

  float* yb = y + (size_t)m0 * D_HID + n0 + l16;
#pragma unroll
  for (int v = 0; v < 8; ++v) {
    float* o = yb + (size_t)(v + 8 * half) * D_HID;
#pragma unroll
    for (int j = 0; j < 4; ++j) o[j * 16] = acc[j][v];
  }
}

// ---------------------------------------------------------------------------
// GEMM2: z[N,128] = relu(p1[N,256] + b1) @ W2[256,128], fp32 WMMA 16x16x4.
// Activation + bias fused into the A-fragment load.
// One wave: 16-row tile x all 8 col-tiles (full 128 cols).
// ---------------------------------------------------------------------------
__global__ void gemm_hw2_kernel(const float* __restrict__ p1,
                                const float* __restrict__ b1,
                                const float* __restrict__ W2,
                                float* __restrict__ z, int rowTiles) {
  const int wave = (blockIdx.x * blockDim.x + threadIdx.x) >> 5;
  const int lane = threadIdx.x & 31;
  if (wave >= rowTiles) return;

  const int m0   = wave * 16;
  const int half = lane >> 4;
  const int l16  = lane & 15;

  v8f acc[8] = {};
  const float* prow = p1 + (size_t)(m0 + l16) * D_HID + 2 * half;

  for (int k = 0; k < D_HID; k += 4) {
    if (k + 64 < D_HID) __builtin_prefetch(prow + k + 64, 0, 1);
    v2f a;
    a.x = fmaxf(prow[k]     + b1[k + 2 * half],     0.0f);
    a.y = fmaxf(prow[k + 1] + b1[k + 2 * half + 1], 0.0f);
    const float* bp = W2 + (size_t)(k + 2 * half) * D_OUT + l16;
#pragma unroll
    for (int j = 0; j < 8; ++j) {
      v2f b;
      b.x = bp[j * 16];
      b.y = bp[j * 16 + D_OUT];
      acc[j] = __builtin_amdgcn_wmma_f32_16x16x4_f32(
          false, a, false, b, (short)0, acc[j], false, false);
    }
  }

  float* zb = z + (size_t)m0 * D_OUT + l16;
#pragma unroll
  for (int v = 0; v < 8; ++v) {
    float* o = zb + (size_t)(v + 8 * half) * D_OUT;
#pragma unroll
    for (int j = 0; j < 8; ++j) o[j * 16] = acc[j][v];
  }
}

// ---------------------------------------------------------------------------
// Self-loop init: out[i,:] = feat[i,:] * dinv[i]^2   (D cols, float4/lane)
// ---------------------------------------------------------------------------
template <int D>
__global__ void selfloop_init_kernel(const float* __restrict__ feat,
                                     const float* __restrict__ dinv,
                                     float* __restrict__ out, int n4) {
  int i = blockIdx.x * blockDim.x + threadIdx.x;
  if (i >= n4) return;
  int node = i / (D / 4);
  float s = dinv[node];
  s *= s;
  float4 v = ((const float4*)feat)[i];
  v.x *= s; v.y *= s; v.z *= s; v.w *= s;
  ((float4*)out)[i] = v;
}

// out[i,:] = b2 + z[i,:] * dinv[i]^2   (D_OUT=128)
__global__ void out_init_kernel(const float* __restrict__ z,
                                const float* __restrict__ dinv,
                                const float* __restrict__ b2,
                                float* __restrict__ out, int n4) {
  int i = blockIdx.x * blockDim.x + threadIdx.x;
  if (i >= n4) return;
  int node = i >> 5;          // 32 float4 per node (128 cols)
  int c4   = i & 31;
  float s = dinv[node];
  s *= s;
  float4 v  = ((const float4*)z)[i];
  float4 bb = ((const float4*)b2)[c4];
  float4 r;
  r.x = v.x * s + bb.x;
  r.y = v.y * s + bb.y;
  r.z = v.z * s + bb.z;
  r.w = v.w * s + bb.w;
  ((float4*)out)[i] = r;
}

// ---------------------------------------------------------------------------
// Edge scatter: out[dst,:] += feat[src,:] * dinv[src]*dinv[dst]
// D/4 lanes per edge (float4 gather, 4 f32 atomics scatter -> L2).
// ---------------------------------------------------------------------------
template <int D>
__global__ void edge_scatter_kernel(const float* __restrict__ feat,
                                    const int* __restrict__ src,
                                    const int* __restrict__ dst,
                                    const float* __restrict__ dinv,
                                    float* __restrict__ out, int E) {
  constexpr int LPE = D / 4;       // lanes per edge
  constexpr int EPB = 256 / LPE;   // edges per 256-thread block
  const int le = threadIdx.x % LPE;
  const int ei = threadIdx.x / LPE;
  const int e  = blockIdx.x * EPB + ei;
  if (e >= E) return;

  const int s = src[e];
  const int d = dst[e];
  const float norm = dinv[s] * dinv[d];

  const float4 v = ((const float4*)(feat + (size_t)s * D))[le];
  float* o = out + (size_t)d * D + le * 4;
  atomicAdd(o + 0, v.x * norm);
  atomicAdd(o + 1, v.y * norm);
  atomicAdd(o + 2, v.z * norm);
  atomicAdd(o + 3, v.w * norm);
}

// ---------------------------------------------------------------------------
// Launch
// ---------------------------------------------------------------------------
extern "C" void kernel_launch(void* const* d_in, const int* in_sizes, int n_in,
                              void* d_out, int out_size, void* d_ws, size_t ws_size,
                              hipStream_t stream) {
  const float* x   = (const float*)d_in[0];
  const int*   ei  = (const int*)d_in[1];
  const float* W1  = (const float*)d_in[2];
  const float* b1  = (const float*)d_in[3];
  const float* W2  = (const float*)d_in[4];
  const float* b2  = (const float*)d_in[5];
  float*       out = (float*)d_out;

  const int N = in_sizes[0] / D_IN;      // 50000
  const int E = in_sizes[1] / 2;         // 1600000
  const int* src = ei;
  const int* dst = ei + E;

  // Workspace layout
  char* ws = (char*)d_ws;
  float* dinv = (float*)ws;                                   // N
  float* y    = (float*)(ws + (((size_t)N * 4 + 255) & ~255ull));
  float* p1   = y + (size_t)N * D_HID;                        // N*256
  float* z    = p1 + (size_t)N * D_HID;                       // N*128

  const int rowTiles = N / 16;  // 3125 (N divisible by 16)

  // 1) degree + rsqrt
  deg_init_kernel<<<(N + 255) / 256, 256, 0, stream>>>(dinv, N);
  deg_scatter_kernel<<<(E + 255) / 256, 256, 0, stream>>>(dst, dinv, E);
  dinv_kernel<<<(N + 255) / 256, 256, 0, stream>>>(dinv, N);

  // 2) y = x @ W1   (fp32 WMMA)
  {
    int waves = rowTiles * (D_HID / 64);
    gemm_xw1_kernel<<<(waves + 7) / 8, 256, 0, stream>>>(x, W1, y, rowTiles);
  }

  // 3) p1 = prop(y): self-loop init + edge scatter
  {
    int n4 = N * (D_HID / 4);
    selfloop_init_kernel<D_HID><<<(n4 + 255) / 256, 256, 0, stream>>>(y, dinv, p1, n4);
    int epb = 256 / (D_HID / 4);  // 4 edges / block
    edge_scatter_kernel<D_HID><<<(E + epb - 1) / epb, 256, 0, stream>>>(
        y, src, dst, dinv, p1, E);
  }

  // 4) z = relu(p1 + b1) @ W2   (fp32 WMMA, fused activation)
  gemm_hw2_kernel<<<(rowTiles + 7) / 8, 256, 0, stream>>>(p1, b1, W2, z, rowTiles);

  // 5) out = prop(z) + b2: init with bias + self-loop, then edge scatter
  {
    int n4 = N * (D_OUT / 4);
    out_init_kernel<<<(n4 + 255) / 256, 256, 0, stream>>>(z, dinv, b2, out, n4);
    int epb = 256 / (D_OUT / 4);  // 8 edges / block
    edge_scatter_kernel<D_OUT><<<(E + epb - 1) / epb, 256, 0, stream>>>(
        z, src, dst, dinv, out, E);
  }
}